// GCNAutoencoder_86114094285160
// MI455X (gfx1250) — compile-verified
//
#include <hip/hip_runtime.h>
#include <cmath>

typedef __attribute__((ext_vector_type(2))) float v2f;
typedef __attribute__((ext_vector_type(8))) float v8f;

constexpr int   N_NODES = 16384;
constexpr int   IN_DIM  = 256;
constexpr int   HID     = 128;
constexpr int   EMB     = 64;
constexpr int   NEDGE   = 524288;

// D = A(16x4 f32) * B(4x16 f32) + C(16x16 f32), wave32 striped.
__device__ __forceinline__ v8f wmma4(v2f a, v2f b, v8f c) {
  return __builtin_amdgcn_wmma_f32_16x16x4_f32(
      /*neg_a=*/false, a, /*neg_b=*/false, b,
      /*c_mod=*/(short)0, c, /*reuse_a=*/false, /*reuse_b=*/false);
}

__device__ __forceinline__ float softplus_f(float x) {
  // stable: max(x,0) + log1p(exp(-|x|))
  return fmaxf(x, 0.0f) + log1pf(__expf(-fabsf(x)));
}

// non-returning fast f32 atomic add -> global_atomic_add_f32 (device scope)
__device__ __forceinline__ void fatomic_add(float* p, float v) {
  __hip_atomic_fetch_add(p, v, __ATOMIC_RELAXED, __HIP_MEMORY_SCOPE_AGENT);
}

// ---------------------------------------------------------------- utilities
__global__ void fill_kernel(float* __restrict__ p, float v, int n) {
  int i = blockIdx.x * 256 + threadIdx.x;
  if (i < n) p[i] = v;
}

// deg[col] += w  (self-loop contribution of +1 is baked into the fill)
__global__ void degree_kernel(const int* __restrict__ ei,
                              const float* __restrict__ ew,
                              float* __restrict__ deg) {
  int e = blockIdx.x * 256 + threadIdx.x;
  if (e < NEDGE) fatomic_add(&deg[ei[NEDGE + e]], ew[e]);
}

__global__ void dinv_kernel(float* __restrict__ deg, int n) {
  int i = blockIdx.x * 256 + threadIdx.x;
  if (i < n) {
    float d = deg[i];
    deg[i] = (d > 0.0f) ? rsqrtf(d) : 0.0f;
  }
}

template <int F, bool RELU>
__global__ void bias_act_kernel(float* __restrict__ p,
                                const float* __restrict__ b, int n) {
  int i = blockIdx.x * 256 + threadIdx.x;
  if (i >= n) return;
  float v = p[i] + b[i & (F - 1)];
  p[i] = RELU ? fmaxf(v, 0.0f) : v;
}

// ------------------------------------------------- normalized scatter-add
// One wave32 per (edge | self-loop).  out[col] += dinv[row]*w*dinv[col]*h[row]
template <int F>
__global__ void agg_kernel(const int* __restrict__ ei,
                           const float* __restrict__ ew,
                           const float* __restrict__ dinv,
                           const float* __restrict__ h,
                           float* __restrict__ out) {
  int g    = blockIdx.x * 8 + (threadIdx.x >> 5);
  int lane = threadIdx.x & 31;
  if (g >= NEDGE + N_NODES) return;
  int r, c;
  float w;
  if (g < NEDGE) {
    r = ei[g];
    c = ei[NEDGE + g];
    w = ew[g];
  } else {
    r = c = g - NEDGE;
    w = 1.0f;
  }
  constexpr int V = F / 32;  // floats per lane
  const float* hp = h + (size_t)r * F + lane * V;
  __builtin_prefetch(hp, 0, 3);  // global_prefetch of the gathered row
  float norm = dinv[r] * w * dinv[c];
  float* op = out + (size_t)c * F + lane * V;
#pragma unroll
  for (int j = 0; j < V; ++j) fatomic_add(&op[j], norm * hp[j]);
}

// --------------------------------------------------------- WMMA dense GEMM
// C[M, NCOLS] = A[M, K] * B[K, NCOLS]; M multiple of 128, K multiple of 64.
// Block: 256 threads = 8 waves; block tile = 128 rows x NCOLS; wave w owns
// 16-row strip, iterates over NCOLS/16 column subtiles. K staged in 64-chunks.
template <int NCOLS>
__global__ void __launch_bounds__(256)
gemm_kernel(const float* __restrict__ A, const float* __restrict__ B,
            float* __restrict__ C, int K) {
  __shared__ float As[128][66];     // pad: stride 264B, 8B aligned -> b64 LDS loads
  __shared__ float Bs[64][NCOLS];
  const int t   = threadIdx.x;
  const int m0  = blockIdx.x * 128;
  const int w   = t >> 5;
  const int l   = t & 31;
  const int hm  = l & 15;
  const int grp = l >> 4;
  constexpr int NT = NCOLS / 16;

  v8f acc[NT];
#pragma unroll
  for (int nt = 0; nt < NT; ++nt)
#pragma unroll
    for (int r = 0; r < 8; ++r) acc[nt][r] = 0.0f;

  for (int kc = 0; kc < K; kc += 64) {
    // stage A chunk: 128 x 64
    for (int i = t; i < 128 * 16; i += 256) {
      int r = i >> 4, c = (i & 15) << 2;
      float4 v = *(const float4*)(A + (size_t)(m0 + r) * K + kc + c);
      As[r][c + 0] = v.x; As[r][c + 1] = v.y;
      As[r][c + 2] = v.z; As[r][c + 3] = v.w;
    }
    // stage B chunk: 64 x NCOLS
    for (int i = t; i < 64 * (NCOLS / 4); i += 256) {
      int r = i / (NCOLS / 4), c = (i % (NCOLS / 4)) << 2;
      float4 v = *(const float4*)(B + (size_t)(kc + r) * NCOLS + c);
      Bs[r][c + 0] = v.x; Bs[r][c + 1] = v.y;
      Bs[r][c + 2] = v.z; Bs[r][c + 3] = v.w;
    }
    __syncthreads();

    v2f av[16];
#pragma unroll
    for (int s = 0; s < 16; ++s)
      av[s] = *(const v2f*)&As[w * 16 + hm][s * 4 + grp * 2];

#pragma unroll
    for (int nt = 0; nt < NT; ++nt) {
#pragma unroll
      for (int s = 0; s < 16; ++s) {
        v2f b;
        b.x = Bs[s * 4 + grp * 2 + 0][nt * 16 + hm];
        b.y = Bs[s * 4 + grp * 2 + 1][nt * 16 + hm];
        acc[nt] = wmma4(av[s], b, acc[nt]);
      }
    }
    __syncthreads();
  }

#pragma unroll
  for (int nt = 0; nt < NT; ++nt)
#pragma unroll
    for (int r = 0; r < 8; ++r) {
      int m = m0 + w * 16 + r + grp * 8;  // C layout: lanes 16-31 hold M=r+8
      int n = nt * 16 + hm;
      C[(size_t)m * NCOLS + n] = acc[nt][r];
    }
}

// ------------------------------------------------ decoder: softplus(z z^T)
// Block = 256 threads (8 waves), 128x128 output tile. z tiles in LDS.
// Two independent WMMA accumulator chains per subtile (even/odd k-steps)
// halve the C->D dependency depth of the 16-deep K loop.
__global__ void __launch_bounds__(256)
decoder_kernel(const float* __restrict__ z, const float* __restrict__ dec_bias,
               float* __restrict__ out) {
  __shared__ float Zm[128][66];
  __shared__ float Zn[128][66];
  const int t  = threadIdx.x;
  const int n0 = blockIdx.x * 128;
  const int m0 = blockIdx.y * 128;

  for (int i = t; i < 128 * 16; i += 256) {  // 128 rows x 16 float4
    int r = i >> 4, c = (i & 15) << 2;
    float4 vm = *(const float4*)(z + (size_t)(m0 + r) * EMB + c);
    Zm[r][c + 0] = vm.x; Zm[r][c + 1] = vm.y;
    Zm[r][c + 2] = vm.z; Zm[r][c + 3] = vm.w;
    float4 vn = *(const float4*)(z + (size_t)(n0 + r) * EMB + c);
    Zn[r][c + 0] = vn.x; Zn[r][c + 1] = vn.y;
    Zn[r][c + 2] = vn.z; Zn[r][c + 3] = vn.w;
  }
  __syncthreads();

  const int w    = t >> 5;
  const int l    = t & 31;
  const int hm   = l & 15;
  const int grp  = l >> 4;
  const float bv = dec_bias[0];

  // A fragments for this wave's 16-row strip (K = 64 -> 16 WMMA steps)
  v2f av[16];
#pragma unroll
  for (int s = 0; s < 16; ++s)
    av[s] = *(const v2f*)&Zm[w * 16 + hm][s * 4 + grp * 2];

  for (int nt = 0; nt < 8; ++nt) {
    v8f acc0, acc1;
#pragma unroll
    for (int r = 0; r < 8; ++r) { acc0[r] = 0.0f; acc1[r] = 0.0f; }
#pragma unroll
    for (int s = 0; s < 8; ++s) {
      // B[k][n] = z[n0+16*nt+n][k]  -> same striped layout as A
      v2f b0 = *(const v2f*)&Zn[nt * 16 + hm][(2 * s + 0) * 4 + grp * 2];
      v2f b1 = *(const v2f*)&Zn[nt * 16 + hm][(2 * s + 1) * 4 + grp * 2];
      acc0 = wmma4(av[2 * s + 0], b0, acc0);
      acc1 = wmma4(av[2 * s + 1], b1, acc1);
    }
#pragma unroll
    for (int r = 0; r < 8; ++r) {
      int m = m0 + w * 16 + r + grp * 8;
      int n = n0 + nt * 16 + hm;
      out[(size_t)m * N_NODES + n] = softplus_f(acc0[r] + acc1[r] + bv);
    }
  }
}

// --------------------------------------------------------------- launcher
extern "C" void kernel_launch(void* const* d_in, const int* in_sizes, int n_in,
                              void* d_out, int out_size, void* d_ws,
                              size_t ws_size, hipStream_t stream) {
  const float* x     = (const float*)d_in[0];
  const int*   ei    = (const int*)d_in[1];
  const float* ew    = (const float*)d_in[2];
  const float* W1    = (const float*)d_in[3];
  const float* b1    = (const float*)d_in[4];
  const float* W2    = (const float*)d_in[5];
  const float* b2    = (const float*)d_in[6];
  const float* dbias = (const float*)d_in[7];

  float* out  = (float*)d_out;
  float* zbuf = out + (size_t)N_NODES * N_NODES;  // z output slot, also decoder input

  float* deg = (float*)d_ws;                      // N     (becomes dinv)
  float* h1  = deg + N_NODES;                     // N*128 (x @ W1)
  float* ag1 = h1 + (size_t)N_NODES * HID;        // N*128 (aggregated, relu'd)
  float* h2  = ag1 + (size_t)N_NODES * HID;       // N*64  (h @ W2)

  auto cdiv = [](int a, int b) { return (a + b - 1) / b; };

  // degrees (self-loop weight 1 baked into init), then deg -> deg^{-1/2}
  fill_kernel<<<cdiv(N_NODES, 256), 256, 0, stream>>>(deg, 1.0f, N_NODES);
  fill_kernel<<<cdiv(N_NODES * HID, 256), 256, 0, stream>>>(ag1, 0.0f, N_NODES * HID);
  fill_kernel<<<cdiv(N_NODES * EMB, 256), 256, 0, stream>>>(zbuf, 0.0f, N_NODES * EMB);
  degree_kernel<<<cdiv(NEDGE, 256), 256, 0, stream>>>(ei, ew, deg);
  dinv_kernel<<<cdiv(N_NODES, 256), 256, 0, stream>>>(deg, N_NODES);

  // layer 1: h1 = x @ W1 ; aggregate ; + b1 ; relu
  gemm_kernel<HID><<<N_NODES / 128, 256, 0, stream>>>(x, W1, h1, IN_DIM);
  agg_kernel<HID><<<cdiv(NEDGE + N_NODES, 8), 256, 0, stream>>>(ei, ew, deg, h1, ag1);
  bias_act_kernel<HID, true>
      <<<cdiv(N_NODES * HID, 256), 256, 0, stream>>>(ag1, b1, N_NODES * HID);

  // layer 2: h2 = h @ W2 ; aggregate into z slot ; + b2
  gemm_kernel<EMB><<<N_NODES / 128, 256, 0, stream>>>(ag1, W2, h2, HID);
  agg_kernel<EMB><<<cdiv(NEDGE + N_NODES, 8), 256, 0, stream>>>(ei, ew, deg, h2, zbuf);
  bias_act_kernel<EMB, false>
      <<<cdiv(N_NODES * EMB, 256), 256, 0, stream>>>(zbuf, b2, N_NODES * EMB);

  // decoder: out = softplus(z @ z^T + bias)
  dim3 dgrid(N_NODES / 128, N_NODES / 128);
  decoder_kernel<<<dgrid, 256, 0, stream>>>(zbuf, dbias, out);
}